// HyperSageLayer_69672959476357
// MI455X (gfx1250) — compile-verified
//
#include <hip/hip_runtime.h>

// Problem constants (from the reference)
#define B_DIM 2
#define V_DIM 50000
#define E_DIM 10000
#define Q_DIM 32
#define F_DIM 256
#define O_DIM 256
#define LDS_STRIDE 260   // 16x256 fp32 A-panel, padded: 260 % 64 = 4 -> conflict-free row stride

typedef __attribute__((ext_vector_type(2))) float v2f;
typedef __attribute__((ext_vector_type(8))) float v8f;

__global__ __launch_bounds__(256) void zero_kernel(float* __restrict__ p, int n) {
    int i = blockIdx.x * blockDim.x + threadIdx.x;
    if (i < n) p[i] = 0.0f;
}

// deg[v] += 1 for every occurrence in edge_nodes (E*Q entries)
__global__ __launch_bounds__(256) void deg_kernel(const int* __restrict__ en,
                                                  float* __restrict__ deg) {
    int i = blockIdx.x * blockDim.x + threadIdx.x;
    if (i < E_DIM * Q_DIM) atomicAdd(&deg[en[i]], 1.0f);
}

// agg[b,e,f] = mean_q X[b, edge_nodes[e,q], f]; one block per (b,e), thread f = feature
__global__ __launch_bounds__(256) void agg_kernel(const float* __restrict__ X,
                                                  const int* __restrict__ en,
                                                  float* __restrict__ agg) {
    __shared__ int sidx[Q_DIM];
    const int be = blockIdx.x;
    const int b  = be / E_DIM;
    const int e  = be % E_DIM;
    const int f  = threadIdx.x;
    if (threadIdx.x < Q_DIM) sidx[threadIdx.x] = en[e * Q_DIM + threadIdx.x];
    __syncthreads();
    float s = 0.0f;
#pragma unroll 8
    for (int q = 0; q < Q_DIM; ++q)
        s += X[((size_t)b * V_DIM + sidx[q]) * F_DIM + f];
    agg[(size_t)be * F_DIM + f] = s * (1.0f / (float)Q_DIM);
}

// acc[b, edge_nodes[e,q], f] += agg[b,e,f]; one block per flat (e,q)
__global__ __launch_bounds__(256) void scatter_kernel(const int* __restrict__ en,
                                                      const float* __restrict__ agg,
                                                      float* __restrict__ acc) {
    const int eq = blockIdx.x;
    const int e  = eq / Q_DIM;
    const int v  = en[eq];
    const int f  = threadIdx.x;
#pragma unroll
    for (int b = 0; b < B_DIM; ++b)
        atomicAdd(&acc[((size_t)b * V_DIM + v) * F_DIM + f],
                  agg[((size_t)b * E_DIM + e) * F_DIM + f]);
}

// Y = (acc / max(deg,1)) @ W^T + bias, computed IN PLACE on `io` (F_DIM == O_DIM).
// Safe: each block stages its 16 rows into LDS before overwriting them; no row is
// shared across blocks. 8 wave32s per block; wave owns two 16x16 N-tiles; K stepped
// by 4 via V_WMMA_F32_16X16X4_F32 (full fp32 precision, matches reference dtype).
__global__ __launch_bounds__(256) void gemm_kernel(float* __restrict__ io,
                                                   const float* __restrict__ deg,
                                                   const float* __restrict__ W,
                                                   const float* __restrict__ bias) {
    __shared__ float lds_a[16 * LDS_STRIDE];
    const int tid     = threadIdx.x;
    const int rowBase = blockIdx.x * 16;   // row in flattened (B*V) space

    // Stage 16x256 panel, folding in 1/deg normalization (coalesced: thread = column)
#pragma unroll
    for (int i = 0; i < 16; ++i) {
        const int grow  = rowBase + i;
        const int v     = grow % V_DIM;
        const float scl = 1.0f / fmaxf(deg[v], 1.0f);
        lds_a[i * LDS_STRIDE + tid] = io[(size_t)grow * F_DIM + tid] * scl;
    }
    __syncthreads();

    const int wave   = tid >> 5;
    const int lane   = tid & 31;
    const int m_lane = lane & 15;          // A row / C column index within tile
    const int k_half = (lane >> 4) << 1;   // lanes 0-15 -> K{0,1}, lanes 16-31 -> K{2,3}

#pragma unroll
    for (int nt = 0; nt < 2; ++nt) {
        const int n = ((wave << 1) + nt) * 16 + m_lane;  // output column this lane owns
        const float* wrow = &W[(size_t)n * F_DIM];       // B[k][n] = W[n][k]
        v8f c = {};
#pragma unroll 8
        for (int k0 = 0; k0 < F_DIM; k0 += 4) {
            // A 16x4 (ISA 32-bit A layout): VGPR j holds K = k_half + j for row m_lane
            v2f a = *(const v2f*)&lds_a[m_lane * LDS_STRIDE + k0 + k_half];
            // B 4x16: VGPR j holds B[K=k_half+j][N = lane%16]
            v2f bm = *(const v2f*)&wrow[k0 + k_half];
            c = __builtin_amdgcn_wmma_f32_16x16x4_f32(
                    /*neg_a=*/false, a, /*neg_b=*/false, bm,
                    /*c_mod=*/(short)0, c, /*reuse_a=*/false, /*reuse_b=*/false);
        }
        // C/D layout: VGPR r, lane l -> M = r + 8*(l/16), N = l%16
        const int mrow = (lane >> 4) << 3;
        const float bv = bias[n];
#pragma unroll
        for (int r = 0; r < 8; ++r)
            io[(size_t)(rowBase + mrow + r) * O_DIM + n] = c[r] + bv;
    }
}

extern "C" void kernel_launch(void* const* d_in, const int* in_sizes, int n_in,
                              void* d_out, int out_size, void* d_ws, size_t ws_size,
                              hipStream_t stream) {
    const float* X    = (const float*)d_in[0];   // (B, V, F)
    const int*   en   = (const int*)d_in[1];     // (E, Q) indices (int per harness convention)
    const float* W    = (const float*)d_in[2];   // (O, F)
    const float* bias = (const float*)d_in[3];   // (O,)
    float*       out  = (float*)d_out;           // (B, V, O); doubles as scatter accumulator

    // Workspace: deg (V floats) | agg (B*E*F floats)  -> ~20.7 MB total
    char*  ws  = (char*)d_ws;
    float* deg = (float*)ws;
    float* agg = (float*)(ws + ((V_DIM * sizeof(float) + 255) & ~(size_t)255));

    zero_kernel<<<(V_DIM + 255) / 256, 256, 0, stream>>>(deg, V_DIM);
    const int accN = B_DIM * V_DIM * F_DIM;
    zero_kernel<<<(accN + 255) / 256, 256, 0, stream>>>(out, accN);

    deg_kernel<<<(E_DIM * Q_DIM + 255) / 256, 256, 0, stream>>>(en, deg);
    agg_kernel<<<B_DIM * E_DIM, 256, 0, stream>>>(X, en, agg);
    scatter_kernel<<<E_DIM * Q_DIM, 256, 0, stream>>>(en, agg, out);
    gemm_kernel<<<(B_DIM * V_DIM) / 16, 256, 0, stream>>>(out, deg, W, bias);
}